// MoleculeGCN_3375844294866
// MI455X (gfx1250) — compile-verified
//
#include <hip/hip_runtime.h>
#include <hip/hip_bf16.h>

#define NNODES 74240
#define NEDGES 593920
#define DIN    75
#define DD     128
#define NLAYER 3
#define BN_EPS 1e-5f

typedef __attribute__((ext_vector_type(2))) float v2f;
typedef __attribute__((ext_vector_type(8))) float v8f;

// ---------------------------------------------------------------------------
// V_WMMA_F32_16X16X4_F32: D(16x16,f32) = A(16x4,f32) * B(4x16,f32) + C
// A frag: VGPR0 = K (lanes0-15: k0, lanes16-31: k0+2), VGPR1 = K+1
// B frag: same K striping, N = lane&15
// C/D:    VGPR r -> row r (lanes0-15) / row r+8 (lanes16-31), N = lane&15
// ---------------------------------------------------------------------------
__device__ __forceinline__ v8f wmma_f32x4(v2f a, v2f b, v8f c) {
    return __builtin_amdgcn_wmma_f32_16x16x4_f32(
        /*neg_a=*/false, a, /*neg_b=*/false, b,
        /*c_mod=*/(short)0, c, /*reuse_a=*/false, /*reuse_b=*/false);
}

// ---------------------------------------------------------------------------
// h0 = X[N,75] @ Winit[75,128]  (K padded to 76; tail step handled branch-free)
// 256 threads = 8 waves; each wave -> one 16-row strip x all 128 cols.
// ---------------------------------------------------------------------------
__global__ void gcn_gemm_init(const float* __restrict__ X,
                              const float* __restrict__ W,
                              float* __restrict__ H) {
    __shared__ float Wl[76 * DD];                 // 38.9 KB, row 75 zeroed
    const int t = threadIdx.x;
    {
        const float4* W4  = (const float4*)W;     // 75*128/4 = 2400 float4
        float4*       Wl4 = (float4*)Wl;
        for (int i = t; i < 2400; i += 256) Wl4[i] = W4[i];
        if (t < 32) Wl4[2400 + t] = make_float4(0.f, 0.f, 0.f, 0.f);
    }
    __syncthreads();

    const int wave  = t >> 5;
    const int lane  = t & 31;
    const int m0    = blockIdx.x * 128 + wave * 16;
    const int row   = m0 + (lane & 15);
    const int khalf = (lane >> 4) << 1;           // 0 or 2
    const int cl    = lane & 15;

    v8f acc[8];
#pragma unroll
    for (int i = 0; i < 8; ++i) acc[i] = {};

    const float* Arow = X + (size_t)row * DIN;
#pragma unroll 1
    for (int k0 = 0; k0 < 72; k0 += 4) {          // k = 0..71 always in-bounds
        const int ka = k0 + khalf;
        v2f a = { Arow[ka], Arow[ka + 1] };
#pragma unroll
        for (int nt = 0; nt < 8; ++nt) {
            const int col = nt * 16 + cl;
            v2f b = { Wl[(k0 + khalf) * DD + col],
                      Wl[(k0 + khalf + 1) * DD + col] };
            acc[nt] = wmma_f32x4(a, b, acc[nt]);
        }
    }
    {   // tail k0 = 72: lanes 0-15 read k=72,73; lanes 16-31 read k=74, pad 0
        const int ka = 72 + khalf;                // 72 or 74, both valid
        v2f a;
        a.x = Arow[ka];
        a.y = (khalf == 0) ? Arow[73] : 0.0f;     // k=75 is the zero pad
#pragma unroll
        for (int nt = 0; nt < 8; ++nt) {
            const int col = nt * 16 + cl;
            v2f b = { Wl[(72 + khalf) * DD + col],
                      Wl[(72 + khalf + 1) * DD + col] };   // row 75 = zeros
            acc[nt] = wmma_f32x4(a, b, acc[nt]);
        }
    }

    const int rbase = m0 + ((lane >> 4) << 3);
#pragma unroll
    for (int nt = 0; nt < 8; ++nt)
#pragma unroll
        for (int r = 0; r < 8; ++r)
            H[(size_t)(rbase + r) * DD + nt * 16 + cl] = acc[nt][r];
}

// ---------------------------------------------------------------------------
// agg[dst[e]] += normalize(h_raw[src[e]])   (BN affine fused into the gather)
// One wave32 per edge; float4 per lane; hardware global_atomic_add_f32.
// ---------------------------------------------------------------------------
__global__ void gcn_scatter(const float* __restrict__ H,
                            const float* __restrict__ ss,
                            const int* __restrict__ src,
                            const int* __restrict__ dst,
                            float* __restrict__ Agg) {
    const int e    = blockIdx.x * (blockDim.x >> 5) + (threadIdx.x >> 5);
    const int lane = threadIdx.x & 31;
    const int c    = lane * 4;
    const float4 sc = *(const float4*)(ss + c);
    const float4 sh = *(const float4*)(ss + DD + c);
    const int s = src[e];
    const int d = dst[e];
    float4 v = ((const float4*)(H + (size_t)s * DD))[lane];
    v.x = fmaf(v.x, sc.x, sh.x);
    v.y = fmaf(v.y, sc.y, sh.y);
    v.z = fmaf(v.z, sc.z, sh.z);
    v.w = fmaf(v.w, sc.w, sh.w);
    float* p = Agg + (size_t)d * DD + c;
    unsafeAtomicAdd(p + 0, v.x);
    unsafeAtomicAdd(p + 1, v.y);
    unsafeAtomicAdd(p + 2, v.z);
    unsafeAtomicAdd(p + 3, v.w);
}

// ---------------------------------------------------------------------------
// H2 = relu(Agg@Wg + bg) + relu(normalize(Hraw)@Wr + br)
// In-place safe (Hin == H2): each block reads/writes only its own 32 rows,
// with a __syncthreads() separating all A-reads from the epilogue stores.
// 8 waves per block; wave -> 16 rows x 32 cols; Wg/Wr in 32-row LDS K-chunks.
// ---------------------------------------------------------------------------
__global__ void gcn_dual_gemm(const float* __restrict__ Agg,
                              const float* Hin,
                              const float* __restrict__ ss,
                              const float* __restrict__ Wg,
                              const float* __restrict__ Wr,
                              const float* __restrict__ bg,
                              const float* __restrict__ br,
                              float* H2) {
    __shared__ float WgS[32 * DD];               // 16 KB
    __shared__ float WrS[32 * DD];               // 16 KB
    __shared__ float ssS[DD];
    __shared__ float shS[DD];
    const int t     = threadIdx.x;
    const int wave  = t >> 5;
    const int lane  = t & 31;
    const int Mbase = blockIdx.x * 32 + (wave >> 2) * 16;
    const int Nbase = (wave & 3) * 32;
    const int mrow  = Mbase + (lane & 15);
    const int khalf = (lane >> 4) << 1;
    const int c0    = Nbase + (lane & 15);
    const int c1    = c0 + 16;

    if (t < DD)      ssS[t]      = ss[t];
    else             shS[t - DD] = ss[t];

    v8f ag0 = {}, ag1 = {}, ar0 = {}, ar1 = {};

    const float* Ag = Agg + (size_t)mrow * DD;
    const float* Hr = Hin + (size_t)mrow * DD;

#pragma unroll 1
    for (int kc = 0; kc < DD; kc += 32) {
        __syncthreads();
        {
            const float4* Wg4  = (const float4*)(Wg + kc * DD);  // 1024 float4
            const float4* Wr4  = (const float4*)(Wr + kc * DD);
            float4* WgS4 = (float4*)WgS;
            float4* WrS4 = (float4*)WrS;
            for (int i = t; i < 1024; i += 256) {
                WgS4[i] = Wg4[i];
                WrS4[i] = Wr4[i];
            }
        }
        __syncthreads();
#pragma unroll
        for (int k0 = 0; k0 < 32; k0 += 4) {
            const int ka = kc + k0 + khalf;
            const int kb = k0 + khalf;
            v2f a_g = { Ag[ka], Ag[ka + 1] };
            v2f a_r = { fmaf(Hr[ka],     ssS[ka],     shS[ka]),
                        fmaf(Hr[ka + 1], ssS[ka + 1], shS[ka + 1]) };
            v2f bg0 = { WgS[kb * DD + c0], WgS[(kb + 1) * DD + c0] };
            v2f bg1 = { WgS[kb * DD + c1], WgS[(kb + 1) * DD + c1] };
            v2f br0 = { WrS[kb * DD + c0], WrS[(kb + 1) * DD + c0] };
            v2f br1 = { WrS[kb * DD + c1], WrS[(kb + 1) * DD + c1] };
            ag0 = wmma_f32x4(a_g, bg0, ag0);
            ag1 = wmma_f32x4(a_g, bg1, ag1);
            ar0 = wmma_f32x4(a_r, br0, ar0);
            ar1 = wmma_f32x4(a_r, br1, ar1);
        }
    }
    __syncthreads();   // all A-reads done before any in-place store

    const int   rbase = Mbase + ((lane >> 4) << 3);
    const float bgc0 = bg[c0], bgc1 = bg[c1];
    const float brc0 = br[c0], brc1 = br[c1];
#pragma unroll
    for (int r = 0; r < 8; ++r) {
        const size_t ro = (size_t)(rbase + r) * DD;
        H2[ro + c0] = fmaxf(ag0[r] + bgc0, 0.0f) + fmaxf(ar0[r] + brc0, 0.0f);
        H2[ro + c1] = fmaxf(ag1[r] + bgc1, 0.0f) + fmaxf(ar1[r] + brc1, 0.0f);
    }
}

// ---------------------------------------------------------------------------
// BatchNorm statistics over raw h2: per-channel sum / sumsq.
// float4 reads; LDS ds_add_f32 reduction; one global f32 atomic per channel
// per block (512 blocks -> 131K atomics over 256 addresses).
// ---------------------------------------------------------------------------
__global__ void gcn_bn_stats(const float* __restrict__ H2,
                             float* __restrict__ sums) {
    __shared__ float sLDS[DD];
    __shared__ float qLDS[DD];
    const int t = threadIdx.x;                    // 256
    if (t < DD) { sLDS[t] = 0.0f; qLDS[t] = 0.0f; }
    __syncthreads();
    const int c = (t & 31) * 4;
    float4 s = make_float4(0.f, 0.f, 0.f, 0.f);
    float4 q = make_float4(0.f, 0.f, 0.f, 0.f);
    for (int row = blockIdx.x * 8 + (t >> 5); row < NNODES; row += gridDim.x * 8) {
        const float4 v = *(const float4*)(H2 + (size_t)row * DD + c);
        s.x += v.x; s.y += v.y; s.z += v.z; s.w += v.w;
        q.x += v.x * v.x; q.y += v.y * v.y; q.z += v.z * v.z; q.w += v.w * v.w;
    }
    atomicAdd(&sLDS[c + 0], s.x); atomicAdd(&qLDS[c + 0], q.x);
    atomicAdd(&sLDS[c + 1], s.y); atomicAdd(&qLDS[c + 1], q.y);
    atomicAdd(&sLDS[c + 2], s.z); atomicAdd(&qLDS[c + 2], q.z);
    atomicAdd(&sLDS[c + 3], s.w); atomicAdd(&qLDS[c + 3], q.w);
    __syncthreads();
    if (t < DD) {
        unsafeAtomicAdd(&sums[t],      sLDS[t]);
        unsafeAtomicAdd(&sums[DD + t], qLDS[t]);
    }
}

__global__ void gcn_bn_finalize(const float* __restrict__ sums,
                                const float* __restrict__ gamma,
                                const float* __restrict__ beta,
                                float* __restrict__ ss) {
    const int c = threadIdx.x;                    // 128
    const float inv_n = 1.0f / (float)NNODES;
    const float mu  = sums[c] * inv_n;
    const float var = sums[DD + c] * inv_n - mu * mu;
    const float sc  = gamma[c] * rsqrtf(var + BN_EPS);
    ss[c]      = sc;
    ss[DD + c] = beta[c] - mu * sc;
}

// Final-layer only: in-place normalize of d_out (float4 grid-stride).
__global__ void gcn_bn_apply(float* H, const float* __restrict__ ss) {
    __shared__ float sS[2 * DD];
    sS[threadIdx.x] = ss[threadIdx.x];            // blockDim == 256
    __syncthreads();
    float4* H4 = (float4*)H;
    const size_t total4 = (size_t)NNODES * DD / 4;
    const size_t step   = (size_t)gridDim.x * blockDim.x;
    for (size_t i = (size_t)blockIdx.x * blockDim.x + threadIdx.x;
         i < total4; i += step) {
        const int c = (int)((i & 31) * 4);
        float4 v = H4[i];
        v.x = fmaf(v.x, sS[c + 0], sS[DD + c + 0]);
        v.y = fmaf(v.y, sS[c + 1], sS[DD + c + 1]);
        v.z = fmaf(v.z, sS[c + 2], sS[DD + c + 2]);
        v.w = fmaf(v.w, sS[c + 3], sS[DD + c + 3]);
        H4[i] = v;
    }
}

__global__ void gcn_zero4(float4* __restrict__ p, int n4) {
    for (int i = blockIdx.x * blockDim.x + threadIdx.x; i < n4;
         i += gridDim.x * blockDim.x)
        p[i] = make_float4(0.f, 0.f, 0.f, 0.f);
}

__global__ void gcn_ss_identity(float* __restrict__ ss) {
    ss[threadIdx.x] = (threadIdx.x < DD) ? 1.0f : 0.0f;   // 256 threads
}

// ---------------------------------------------------------------------------
// Host launcher. Inputs (setup_inputs order):
// 0 node_feats[N,75] f32, 1 src[E] i32, 2 dst[E] i32, 3 W_init[75,128] f32,
// 4 Wg[3,128,128], 5 bg[3,128], 6 Wr[3,128,128], 7 br[3,128],
// 8 gamma[3,128], 9 beta[3,128].   d_out = h reshaped [256,290,128] f32.
// d_out doubles as the single h/h2 buffer; BN affine is fused into consumers.
// ---------------------------------------------------------------------------
extern "C" void kernel_launch(void* const* d_in, const int* in_sizes, int n_in,
                              void* d_out, int out_size, void* d_ws, size_t ws_size,
                              hipStream_t stream) {
    const float* X     = (const float*)d_in[0];
    const int*   src   = (const int*)  d_in[1];
    const int*   dst   = (const int*)  d_in[2];
    const float* Winit = (const float*)d_in[3];
    const float* Wg    = (const float*)d_in[4];
    const float* bg    = (const float*)d_in[5];
    const float* Wr    = (const float*)d_in[6];
    const float* br    = (const float*)d_in[7];
    const float* gamma = (const float*)d_in[8];
    const float* beta  = (const float*)d_in[9];

    float* H    = (float*)d_out;                      // raw h2 / h buffer
    float* agg  = (float*)d_ws;                       // neighbor sums (38 MB)
    float* sums = agg + (size_t)NNODES * DD;          // [256] sum/sumsq
    float* ss   = sums + 2 * DD;                      // [256] scale/shift

    gcn_ss_identity<<<1, 256, 0, stream>>>(ss);       // layer-0 "h" is h0 as-is
    gcn_gemm_init<<<NNODES / 128, 256, 0, stream>>>(X, Winit, H);

    for (int l = 0; l < NLAYER; ++l) {
        gcn_zero4<<<2048, 256, 0, stream>>>((float4*)agg, NNODES * DD / 4);
        gcn_zero4<<<1, 64, 0, stream>>>((float4*)sums, 2 * DD / 4);
        gcn_scatter<<<NEDGES / 8, 256, 0, stream>>>(H, ss, src, dst, agg);
        gcn_dual_gemm<<<NNODES / 32, 256, 0, stream>>>(
            agg, H, ss, Wg + (size_t)l * DD * DD, Wr + (size_t)l * DD * DD,
            bg + l * DD, br + l * DD, H);
        gcn_bn_stats<<<512, 256, 0, stream>>>(H, sums);
        gcn_bn_finalize<<<1, DD, 0, stream>>>(sums, gamma + l * DD, beta + l * DD, ss);
    }
    gcn_bn_apply<<<2048, 256, 0, stream>>>(H, ss);    // final normalize in place
}